// DeepLinear_37623913513720
// MI455X (gfx1250) — compile-verified
//
#include <hip/hip_runtime.h>

// CDNA5 / gfx1250: wave32, V_WMMA_F32_16X16X4_F32 (A 16x4 f32 -> v2f, B 4x16 -> v2f, C/D 16x16 -> v8f)
typedef __attribute__((ext_vector_type(2))) float v2f;
typedef __attribute__((ext_vector_type(8))) float v8f;

#define BN_EPS 1e-5f
#define BATCH  65536
#define HPAD   16      // h stored padded [B,16], cols 10..15 == 0

// Fragment mapping used (per CDNA5 ISA §7.12.2):
//  A (16x4, MxK): lane l -> M = l&15 ; VGPR v holds K = kk + 2*(l>>4) + v   (v=0,1)
//  B (4x16, KxN): lane l -> N = l&15 ; VGPR v holds K = kk + 2*(l>>4) + v
//  C/D (16x16):   lane l -> N = l&15 ; VGPR r holds M = r + 8*(l>>4)

// ---------------------------------------------------------------------------
// Layer 0: h0 = x @ W0^T + b0 (raw, pre-BN) ; per-block partial sum/sumsq
// grid 512 x 256 threads; wave handles 16 batch rows; block handles 128 rows
// ---------------------------------------------------------------------------
__global__ __launch_bounds__(256) void layer0_kernel(
    const float* __restrict__ x,      // [B,784]
    const float* __restrict__ W0,     // [10,784]
    const float* __restrict__ b0,     // [10]
    float*       __restrict__ hout,   // [B,16] raw h0
    float*       __restrict__ partials) // [gridDim.x][32] (sum[16],sumsq[16])
{
  __shared__ float ldsW[784 * 16];    // W0^T padded: [k][n], 49 KB
  __shared__ float red[8][32];
  const int t = threadIdx.x;

  for (int idx = t; idx < 784 * 16; idx += 256) {
    const int k = idx >> 4, n = idx & 15;
    ldsW[idx] = (n < 10) ? W0[n * 784 + k] : 0.f;
  }
  __syncthreads();

  const int lane  = t & 31, wave = t >> 5;
  const int nsub  = lane & 15;        // A-row offset / B-col / D-col
  const int khalf = lane >> 4;
  const int m0    = blockIdx.x * 128 + wave * 16;

  const float biasn = (nsub < 10) ? b0[nsub] : 0.f;
  v8f c;
  #pragma unroll
  for (int r = 0; r < 8; ++r) c[r] = biasn;

  const float* xrow = x + (size_t)(m0 + nsub) * 784 + 2 * khalf;

  #pragma unroll 4
  for (int kk = 0; kk < 784; kk += 4) {
    v2f a = *(const v2f*)(xrow + kk);                 // 8B aligned
    v2f b;
    b.x = ldsW[(kk + 2 * khalf)     * 16 + nsub];
    b.y = ldsW[(kk + 2 * khalf + 1) * 16 + nsub];
    c = __builtin_amdgcn_wmma_f32_16x16x4_f32(false, a, false, b,
                                              (short)0, c, false, false);
  }

  float sum = 0.f, sumsq = 0.f;
  #pragma unroll
  for (int r = 0; r < 8; ++r) {
    const int row = m0 + r + 8 * khalf;
    hout[(size_t)row * HPAD + nsub] = c[r];
    sum += c[r]; sumsq += c[r] * c[r];
  }
  // lanes l and l+16 hold the same feature column n
  sum   += __shfl_xor(sum,   16, 32);
  sumsq += __shfl_xor(sumsq, 16, 32);
  if (lane < 16) { red[wave][nsub] = sum; red[wave][16 + nsub] = sumsq; }
  __syncthreads();
  if (t < 32) {
    float acc = 0.f;
    #pragma unroll
    for (int w = 0; w < 8; ++w) acc += red[w][t];     // fixed order -> deterministic
    partials[blockIdx.x * 32 + t] = acc;
  }
}

// ---------------------------------------------------------------------------
// Fold BN stats into scale/shift:  hnorm = h*scale + shift
// ---------------------------------------------------------------------------
__global__ __launch_bounds__(64) void reduce_kernel(
    const float* __restrict__ partials, int nblocks,
    const float* __restrict__ g, const float* __restrict__ beta,
    float* __restrict__ stats)  // [32]: scale[16], shift[16]
{
  const int f = threadIdx.x;
  if (f >= 16) return;
  float sum = 0.f, sq = 0.f;
  for (int p = 0; p < nblocks; ++p) {                 // fixed order
    sum += partials[p * 32 + f];
    sq  += partials[p * 32 + 16 + f];
  }
  const float invB = 1.f / (float)BATCH;
  const float mu   = sum * invB;
  const float var  = sq * invB - mu * mu;             // biased, matches torch BN
  const float rstd = rsqrtf(var + BN_EPS);
  const float gg = (f < 10) ? g[f]    : 0.f;
  const float bb = (f < 10) ? beta[f] : 0.f;
  const float sc = gg * rstd;
  stats[f]      = sc;
  stats[16 + f] = bb - mu * sc;                       // pad features -> 0,0
}

// ---------------------------------------------------------------------------
// Mid layer: hnext = BN(hin) @ W^T + b (raw) ; partial stats of hnext
// grid 128 x 256; wave handles 4 M-tiles of 16 rows (block = 512 rows)
// ---------------------------------------------------------------------------
__global__ __launch_bounds__(256) void mid_kernel(
    const float* __restrict__ hin,    // [B,16] raw layer i
    const float* __restrict__ W,      // [10,10]
    const float* __restrict__ bias,   // [10]
    const float* __restrict__ stats,  // scale/shift for BN of layer i
    float*       __restrict__ hout,   // [B,16] raw layer i+1
    float*       __restrict__ partials)
{
  __shared__ float ldsW[16 * 16];     // W^T padded [k][n]
  __shared__ float red[8][32];
  const int t = threadIdx.x;
  {
    const int k = t >> 4, n = t & 15;
    ldsW[t] = (k < 10 && n < 10) ? W[n * 10 + k] : 0.f;
  }
  __syncthreads();

  const int lane  = t & 31, wave = t >> 5;
  const int nsub  = lane & 15, khalf = lane >> 4;
  const float biasn = (nsub < 10) ? bias[nsub] : 0.f;

  v2f bfrag[4];
  float sc0[4], sc1[4], sh0[4], sh1[4];
  #pragma unroll
  for (int s = 0; s < 4; ++s) {
    const int k0 = s * 4 + 2 * khalf;
    bfrag[s].x = ldsW[k0 * 16 + nsub];
    bfrag[s].y = ldsW[(k0 + 1) * 16 + nsub];
    sc0[s] = stats[k0];     sh0[s] = stats[16 + k0];
    sc1[s] = stats[k0 + 1]; sh1[s] = stats[16 + k0 + 1];
  }

  const int m0w = blockIdx.x * 512 + wave * 64;
  float sum = 0.f, sumsq = 0.f;

  #pragma unroll
  for (int tile = 0; tile < 4; ++tile) {
    const int m0 = m0w + tile * 16;
    v8f c;
    #pragma unroll
    for (int r = 0; r < 8; ++r) c[r] = biasn;
    const float* hrow = hin + (size_t)(m0 + nsub) * HPAD + 2 * khalf;
    #pragma unroll
    for (int s = 0; s < 4; ++s) {
      v2f a = *(const v2f*)(hrow + s * 4);
      a.x = a.x * sc0[s] + sh0[s];                    // fused BN on A operand
      a.y = a.y * sc1[s] + sh1[s];
      c = __builtin_amdgcn_wmma_f32_16x16x4_f32(false, a, false, bfrag[s],
                                                (short)0, c, false, false);
    }
    #pragma unroll
    for (int r = 0; r < 8; ++r) {
      const int row = m0 + r + 8 * khalf;
      hout[(size_t)row * HPAD + nsub] = c[r];
      sum += c[r]; sumsq += c[r] * c[r];
    }
  }

  sum   += __shfl_xor(sum,   16, 32);
  sumsq += __shfl_xor(sumsq, 16, 32);
  if (lane < 16) { red[wave][nsub] = sum; red[wave][16 + nsub] = sumsq; }
  __syncthreads();
  if (t < 32) {
    float acc = 0.f;
    #pragma unroll
    for (int w = 0; w < 8; ++w) acc += red[w][t];
    partials[blockIdx.x * 32 + t] = acc;
  }
}

// ---------------------------------------------------------------------------
// Final: out = BN(h62) @ Wf^T + bf  (writes [B,10] exactly)
// ---------------------------------------------------------------------------
__global__ __launch_bounds__(256) void final_kernel(
    const float* __restrict__ hin, const float* __restrict__ Wf,
    const float* __restrict__ bf,  const float* __restrict__ stats,
    float* __restrict__ out)       // [B,10]
{
  __shared__ float ldsW[16 * 16];
  const int t = threadIdx.x;
  {
    const int k = t >> 4, n = t & 15;
    ldsW[t] = (k < 10 && n < 10) ? Wf[n * 10 + k] : 0.f;
  }
  __syncthreads();

  const int lane  = t & 31, wave = t >> 5;
  const int nsub  = lane & 15, khalf = lane >> 4;
  const float biasn = (nsub < 10) ? bf[nsub] : 0.f;

  v2f bfrag[4];
  float sc0[4], sc1[4], sh0[4], sh1[4];
  #pragma unroll
  for (int s = 0; s < 4; ++s) {
    const int k0 = s * 4 + 2 * khalf;
    bfrag[s].x = ldsW[k0 * 16 + nsub];
    bfrag[s].y = ldsW[(k0 + 1) * 16 + nsub];
    sc0[s] = stats[k0];     sh0[s] = stats[16 + k0];
    sc1[s] = stats[k0 + 1]; sh1[s] = stats[16 + k0 + 1];
  }

  const int m0w = blockIdx.x * 512 + wave * 64;
  #pragma unroll
  for (int tile = 0; tile < 4; ++tile) {
    const int m0 = m0w + tile * 16;
    v8f c;
    #pragma unroll
    for (int r = 0; r < 8; ++r) c[r] = biasn;
    const float* hrow = hin + (size_t)(m0 + nsub) * HPAD + 2 * khalf;
    #pragma unroll
    for (int s = 0; s < 4; ++s) {
      v2f a = *(const v2f*)(hrow + s * 4);
      a.x = a.x * sc0[s] + sh0[s];
      a.y = a.y * sc1[s] + sh1[s];
      c = __builtin_amdgcn_wmma_f32_16x16x4_f32(false, a, false, bfrag[s],
                                                (short)0, c, false, false);
    }
    if (nsub < 10) {
      #pragma unroll
      for (int r = 0; r < 8; ++r) {
        const int row = m0 + r + 8 * khalf;
        out[(size_t)row * 10 + nsub] = c[r];
      }
    }
  }
}

// ---------------------------------------------------------------------------
extern "C" void kernel_launch(void* const* d_in, const int* in_sizes, int n_in,
                              void* d_out, int out_size, void* d_ws, size_t ws_size,
                              hipStream_t stream) {
  const float* x     = (const float*)d_in[0];
  const float* W0    = (const float*)d_in[1];
  const float* b0    = (const float*)d_in[2];
  const float* g0    = (const float*)d_in[3];
  const float* beta0 = (const float*)d_in[4];
  const float* Ws    = (const float*)d_in[5];   // [62,10,10]
  const float* bs    = (const float*)d_in[6];   // [62,10]
  const float* gs    = (const float*)d_in[7];   // [62,10]
  const float* betas = (const float*)d_in[8];   // [62,10]
  const float* Wf    = (const float*)d_in[9];
  const float* bf    = (const float*)d_in[10];
  float* out = (float*)d_out;

  float* hA       = (float*)d_ws;               // [65536,16]
  float* hB       = hA + (size_t)BATCH * HPAD;  // [65536,16]
  float* partials = hB + (size_t)BATCH * HPAD;  // [512,32]
  float* stats    = partials + 512 * 32;        // [32]

  layer0_kernel<<<512, 256, 0, stream>>>(x, W0, b0, hA, partials);
  reduce_kernel<<<1, 64, 0, stream>>>(partials, 512, g0, beta0, stats);

  const float* cur = hA;
  float*       nxt = hB;
  for (int i = 0; i < 62; ++i) {
    mid_kernel<<<128, 256, 0, stream>>>(cur, Ws + i * 100, bs + i * 10,
                                        stats, nxt, partials);
    reduce_kernel<<<1, 64, 0, stream>>>(partials, 128, gs + i * 10,
                                        betas + i * 10, stats);
    const float* tmp = nxt; nxt = (float*)cur; cur = tmp;
  }

  final_kernel<<<128, 256, 0, stream>>>(cur, Wf, bf, stats, out);
}